// GetDensity_39144331936466
// MI455X (gfx1250) — compile-verified
//
#include <hip/hip_runtime.h>
#include <math.h>

#define NTYPE 4
#define NWAVE 8
#define NORBIT 64
#define LCH 13            // 1 + 3 + 9 angular channels
#define LW  (LCH*NWAVE)   // 104 floats per atom accumulator row
#define PI_F 3.14159265358979323846f
#define CUTOFF_F 5.0f

typedef __attribute__((ext_vector_type(2))) float v2f;
typedef __attribute__((ext_vector_type(8))) float v8f;
typedef unsigned int u32x4 __attribute__((ext_vector_type(4)));
typedef int          i32x4 __attribute__((ext_vector_type(4)));
typedef int          i32x8 __attribute__((ext_vector_type(8)));

// ---------------------------------------------------------------------------
// TDM bulk stage: copy a 2-D tile (dim0 dwords x dim1 rows, row stride = dim0)
// of f32 from global memory into LDS via the Tensor Data Mover.
// Descriptor layout per CDNA5 ISA (D# groups 0/1).
// Issued by the calling wave only; caller must barrier afterwards.
// ---------------------------------------------------------------------------
__device__ __forceinline__ void tdm_stage_f32(float* lds_dst, const float* gsrc,
                                              unsigned dim0, unsigned dim1,
                                              int tid, int nth)
{
#if __has_builtin(__builtin_amdgcn_tensor_load_to_lds)
    if (tid < 32) {   // one wave issues the DMA (EXEC ignored by TDM)
        unsigned long long ga = (unsigned long long)(uintptr_t)gsrc;
        unsigned lds_addr = (unsigned)(uintptr_t)lds_dst;   // truncate -> LDS byte offset
        u32x4 g0;
        g0[0] = 1u;                                   // count=1, user mode, no gather
        g0[1] = lds_addr;                             // lds_addr [63:32]
        g0[2] = (unsigned)(ga & 0xFFFFFFFFu);         // global_addr[31:0]
        g0[3] = (unsigned)((ga >> 32) & 0x01FFFFFFu)  // global_addr[56:32]
              | (2u << 30);                           // type = 2 ("image")
        i32x8 g1;
        g1[0] = (2 << 16);                            // data_size = 2 (4 bytes)
        g1[1] = (int)((dim0 & 0xFFFFu) << 16);        // tensor_dim0[15:0]
        g1[2] = (int)((dim0 >> 16) & 0xFFFFu)         // tensor_dim0[31:16]
              | (int)((dim1 & 0xFFFFu) << 16);        // tensor_dim1[15:0]
        g1[3] = (int)((dim1 >> 16) & 0xFFFFu)         // tensor_dim1[31:16]
              | (int)((dim0 & 0xFFFFu) << 16);        // tile_dim0
        g1[4] = (int)(dim1 & 0xFFFFu);                // tile_dim1 (tile_dim2 = 0)
        g1[5] = (int)dim0;                            // tensor_dim0_stride[31:0]
        g1[6] = 0;                                    // stride hi / dim1 stride
        g1[7] = 0;
        i32x4 z4 = {0, 0, 0, 0};
#if __clang_major__ >= 23
        i32x8 z8 = {0, 0, 0, 0, 0, 0, 0, 0};
        __builtin_amdgcn_tensor_load_to_lds(g0, g1, z4, z4, z8, 0);
#else
        __builtin_amdgcn_tensor_load_to_lds(g0, g1, z4, z4, 0);
#endif
        __builtin_amdgcn_s_wait_tensorcnt(0);
    }
#else
    for (unsigned i = tid; i < dim0 * dim1; i += nth) lds_dst[i] = gsrc[i];
#endif
}

// ---------------------------------------------------------------------------
// Kernel 1: per-atom init.  ef_orb0[a,l,w] = ang(ef_b)[l] * ef_para[w];
// acc = ef_orb0; accSum = 0; coeff = params_coeff[species].
// ---------------------------------------------------------------------------
__global__ void k_init(const float* __restrict__ ef,
                       const float* __restrict__ ef_para,
                       const float* __restrict__ params_coeff,
                       const int*   __restrict__ species,
                       float* __restrict__ ef_orb0,
                       float* __restrict__ acc,
                       float* __restrict__ accSum,
                       float* __restrict__ coeff,
                       int natom, int tot)
{
    int a = blockIdx.x * blockDim.x + threadIdx.x;
    if (a >= tot) return;
    int b = a / natom;
    float e[3] = { ef[3*b+0], ef[3*b+1], ef[3*b+2] };
    float ang[LCH];
    ang[0] = 1.0f;
    #pragma unroll
    for (int j = 0; j < 3; ++j) ang[1+j] = e[j];
    #pragma unroll
    for (int p = 0; p < 3; ++p)
        #pragma unroll
        for (int j = 0; j < 3; ++j) ang[4+3*p+j] = ang[1+p]*e[j];

    int sp = species[a];
    #pragma unroll
    for (int w = 0; w < NWAVE; ++w) coeff[(size_t)a*NWAVE+w] = params_coeff[sp*NWAVE+w];

    size_t base = (size_t)a*LW;
    #pragma unroll
    for (int l = 0; l < LCH; ++l)
        #pragma unroll
        for (int w = 0; w < NWAVE; ++w) {
            float v = ang[l]*ef_para[w];
            ef_orb0[base + l*NWAVE + w] = v;
            acc    [base + l*NWAVE + w] = v;
            accSum [base + l*NWAVE + w] = 0.0f;
        }
}

// ---------------------------------------------------------------------------
// Per-pair math shared by both scatter kernels: base orbital terms.
// ---------------------------------------------------------------------------
__device__ __forceinline__ bool pair_terms(const float* __restrict__ cart,
                                           const float* __restrict__ shifts,
                                           const int*   __restrict__ species,
                                           const float* __restrict__ rs,
                                           const float* __restrict__ inta,
                                           int p, int g0a, int g1a,
                                           float* ang, float* g, float* dcut_out)
{
    float sx = shifts[3*p+0], sy = shifts[3*p+1], sz = shifts[3*p+2];
    if (!(sx > -1e10f && sy > -1e10f && sz > -1e10f)) return false; // pad -> 0
    float dx = cart[3*g1a+0] - cart[3*g0a+0] + sx;
    float dy = cart[3*g1a+1] - cart[3*g0a+1] + sy;
    float dz = cart[3*g1a+2] - cart[3*g0a+2] + sz;
    float d  = sqrtf(dx*dx + dy*dy + dz*dz);
    float t  = 0.5f*cosf(d*(PI_F/CUTOFF_F)) + 0.5f;
    float dcut = t*t;
    float inv = 1.0f/d;
    float u[3] = { dx*inv, dy*inv, dz*inv };
    int sp = species[g1a];
    #pragma unroll
    for (int w = 0; w < NWAVE; ++w) {
        float diff = d - rs[sp*NWAVE+w];
        g[w] = expf(inta[sp*NWAVE+w]*diff*diff);
    }
    ang[0] = dcut;
    #pragma unroll
    for (int j = 0; j < 3; ++j) ang[1+j] = dcut*u[j];
    #pragma unroll
    for (int q = 0; q < 3; ++q)
        #pragma unroll
        for (int j = 0; j < 3; ++j) ang[4+3*q+j] = ang[1+q]*u[j];
    *dcut_out = dcut;
    return true;
}

// ---------------------------------------------------------------------------
// Kernel 2a: pair scatter with LDS aggregation (requires maxneigh % blockDim
// == 0 so each block sits inside one molecule).  The molecule's accSum tile
// (natom x LW = 53 KB) is staged into LDS by the TDM; per-pair contributions
// aggregate with ds_add_f32 atomics; one global-atomic flush per block.
// ---------------------------------------------------------------------------
__global__ void k_pairs_lds(const float* __restrict__ cart,
                            const int*   __restrict__ atom_index,
                            const float* __restrict__ shifts,
                            const int*   __restrict__ species,
                            const float* __restrict__ rs,
                            const float* __restrict__ inta,
                            const float* __restrict__ coeff,
                            const float* __restrict__ accSum,
                            float* __restrict__ acc,
                            int natom, int maxneigh, int npair, int use_sum)
{
    extern __shared__ float smem[];
    float* lds_acc = smem;                    // natom*LW  scatter accumulator
    float* lds_in  = smem + natom*LW;         // natom*LW  staged accSum
    float* lds_cf  = lds_in + natom*LW;       // natom*NWAVE staged coeff
    const int tid = threadIdx.x, nth = blockDim.x;
    const int p   = blockIdx.x*blockDim.x + tid;
    const int b   = (blockIdx.x*blockDim.x) / maxneigh;   // molecule of this block

    for (int i = tid; i < natom*LW; i += nth) lds_acc[i] = 0.0f;
    for (int i = tid; i < natom*NWAVE; i += nth)
        lds_cf[i] = coeff[(size_t)b*natom*NWAVE + i];
    if (use_sum)
        tdm_stage_f32(lds_in, accSum + (size_t)b*natom*LW, LW, natom, tid, nth);
    __syncthreads();

    if (p < npair) {
        int i0 = atom_index[p];            // local atom ids within molecule
        int i1 = atom_index[npair + p];
        float ang[LCH], g[NWAVE], dcut;
        if (pair_terms(cart, shifts, species, rs, inta, p,
                       b*natom + i0, b*natom + i1, ang, g, &dcut)) {
            const float* cf = lds_cf + i1*NWAVE;
            const float* as = lds_in + i1*LW;
            float*       dl = lds_acc + i0*LW;
            #pragma unroll
            for (int l = 0; l < LCH; ++l)
                #pragma unroll
                for (int w = 0; w < NWAVE; ++w) {
                    float orb = ang[l]*g[w];
                    if (use_sum) orb += dcut*as[l*NWAVE+w];
                    atomicAdd(dl + l*NWAVE + w, orb*cf[w]);     // ds_add_f32
                }
        }
    }
    __syncthreads();

    float* gdst = acc + (size_t)b*natom*LW;
    for (int i = tid; i < natom*LW; i += nth) {
        float v = lds_acc[i];
        if (v != 0.0f) atomicAdd(gdst + i, v);
    }
}

// ---------------------------------------------------------------------------
// Kernel 2b: direct-atomic fallback (generic shapes).
// ---------------------------------------------------------------------------
__global__ void k_pairs_direct(const float* __restrict__ cart,
                               const int*   __restrict__ atom_index,
                               const float* __restrict__ shifts,
                               const int*   __restrict__ species,
                               const float* __restrict__ rs,
                               const float* __restrict__ inta,
                               const float* __restrict__ coeff,
                               const float* __restrict__ accSum,
                               float* __restrict__ acc,
                               int natom, int maxneigh, int npair, int use_sum)
{
    int p = blockIdx.x * blockDim.x + threadIdx.x;
    if (p >= npair) return;
    int b  = p / maxneigh;
    int i0 = atom_index[p]         + b*natom;
    int i1 = atom_index[npair + p] + b*natom;
    float ang[LCH], g[NWAVE], dcut;
    if (!pair_terms(cart, shifts, species, rs, inta, p, i0, i1, ang, g, &dcut))
        return;
    const float* cf = coeff  + (size_t)i1*NWAVE;
    const float* as = accSum + (size_t)i1*LW;
    float*       dst = acc   + (size_t)i0*LW;
    #pragma unroll
    for (int l = 0; l < LCH; ++l)
        #pragma unroll
        for (int w = 0; w < NWAVE; ++w) {
            float orb = ang[l]*g[w];
            if (use_sum) orb += dcut*as[l*NWAVE+w];
            atomicAdd(dst + l*NWAVE + w, orb*cf[w]);
        }
}

// ---------------------------------------------------------------------------
// Kernel 3: per-atom density via f32 WMMA.  One wave per 16-atom tile.
// INDEX_PARA maps the 13 channels onto only 3 distinct hyper matrices, so B
// operands are hoisted per ip-group (loaded once, reused across the group's
// channels).  Per (l, n-tile): C = A(16x4)B(4x16) chained over K=8, then
// density += C*C.
// ---------------------------------------------------------------------------
__global__ void k_density(const float* __restrict__ acc,
                          const float* __restrict__ hyper,   // [3,8,64]
                          float* __restrict__ dst,           // [tot,64]
                          int tot)
{
    int wave = threadIdx.x >> 5;            // 8 waves / block
    int lane = threadIdx.x & 31;
    int tile = blockIdx.x * 8 + wave;
    int base = tile * 16;
    if (base >= tot) return;                // wave-uniform exit

    int m  = lane & 15;                     // M row (and N col for B/D)
    int kp = lane >> 4;                     // K-pair select

    v8f dacc0 = {}, dacc1 = {}, dacc2 = {}, dacc3 = {};

    const int l_lo[3] = {0, 1, 4};          // channel ranges per ip
    const int l_hi[3] = {1, 4, 13};

    #pragma unroll
    for (int ip = 0; ip < 3; ++ip) {
        // Hoist B for this ip: 4 n-tiles x 2 K-halves, v2f each (16 VGPRs).
        const float* hb = hyper + ip*(NWAVE*NORBIT) + (2*kp)*NORBIT;
        v2f b0[4], b1[4];
        #pragma unroll
        for (int nt = 0; nt < 4; ++nt) {
            int n = nt*16 + m;
            b0[nt].x = hb[n];            b0[nt].y = hb[NORBIT + n];      // K=0..3
            b1[nt].x = hb[4*NORBIT + n]; b1[nt].y = hb[5*NORBIT + n];    // K=4..7
        }
        #pragma unroll
        for (int l = l_lo[ip]; l < l_hi[ip]; ++l) {
            const float* arow = acc + (size_t)(base + m)*LW + l*NWAVE + 2*kp;
            v2f a0; a0.x = arow[0]; a0.y = arow[1];        // K = 0..3 half
            v2f a1; a1.x = arow[4]; a1.y = arow[5];        // K = 4..7 half
            #pragma unroll
            for (int nt = 0; nt < 4; ++nt) {
                v8f c = {};
                c = __builtin_amdgcn_wmma_f32_16x16x4_f32(false, a0, false, b0[nt],
                                                          (short)0, c, false, false);
                c = __builtin_amdgcn_wmma_f32_16x16x4_f32(false, a1, false, b1[nt],
                                                          (short)0, c, false, false);
                v8f sq = c*c;
                if      (nt == 0) dacc0 += sq;
                else if (nt == 1) dacc1 += sq;
                else if (nt == 2) dacc2 += sq;
                else              dacc3 += sq;
            }
        }
    }

    // D layout: VGPR j, lanes 0-15 -> M=j, lanes 16-31 -> M=j+8; N = lane&15.
    #pragma unroll
    for (int j = 0; j < 8; ++j) {
        size_t row = (size_t)(base + j + 8*kp)*NORBIT;
        dst[row +  0 + m] = dacc0[j];
        dst[row + 16 + m] = dacc1[j];
        dst[row + 32 + m] = dacc2[j];
        dst[row + 48 + m] = dacc3[j];
    }
}

// ---------------------------------------------------------------------------
// Kernel 4: between-round update.
// coeff += tanh(density @ oc_w[t]) + oc_emb[t][species];
// accSum += acc;  acc = ef_orb0   (ready for next scatter round).
// ---------------------------------------------------------------------------
__global__ void k_update(const float* __restrict__ density,
                         const float* __restrict__ ocw,     // [64,8]
                         const float* __restrict__ ocemb,   // [NTYPE,8]
                         const int*   __restrict__ species,
                         const float* __restrict__ ef_orb0,
                         float* __restrict__ coeff,
                         float* __restrict__ acc,
                         float* __restrict__ accSum,
                         int tot)
{
    int a = blockIdx.x * blockDim.x + threadIdx.x;
    if (a >= tot) return;
    const float* drow = density + (size_t)a*NORBIT;
    int sp = species[a];
    #pragma unroll
    for (int w = 0; w < NWAVE; ++w) {
        float s = 0.0f;
        #pragma unroll
        for (int n = 0; n < NORBIT; ++n) s += drow[n]*ocw[n*NWAVE + w];
        coeff[(size_t)a*NWAVE + w] += tanhf(s) + ocemb[sp*NWAVE + w];
    }
    float*       as = accSum + (size_t)a*LW;
    float*       ac = acc    + (size_t)a*LW;
    const float* e0 = ef_orb0+ (size_t)a*LW;
    #pragma unroll
    for (int i = 0; i < LW; ++i) { as[i] += ac[i]; ac[i] = e0[i]; }
}

// ---------------------------------------------------------------------------
extern "C" void kernel_launch(void* const* d_in, const int* in_sizes, int n_in,
                              void* d_out, int out_size, void* d_ws, size_t ws_size,
                              hipStream_t stream)
{
    const float* cart      = (const float*)d_in[0];
    const float* ef        = (const float*)d_in[1];
    // d_in[2] = numatoms (unused: all natom)
    const int*   species   = (const int*)  d_in[3];
    const int*   atom_idx  = (const int*)  d_in[4];
    const float* shifts    = (const float*)d_in[5];
    const float* rs        = (const float*)d_in[6];
    const float* inta      = (const float*)d_in[7];
    const float* p_coeff   = (const float*)d_in[8];
    const float* hyper     = (const float*)d_in[9];
    const float* ef_para   = (const float*)d_in[10];
    const float* oc_w      = (const float*)d_in[11];
    const float* oc_emb    = (const float*)d_in[12];

    int nbatch   = in_sizes[1] / 3;
    int natom    = in_sizes[0] / (3*nbatch);
    int tot      = nbatch * natom;
    int maxneigh = in_sizes[5] / (3*nbatch);
    int npair    = nbatch * maxneigh;

    float* w        = (float*)d_ws;
    float* ef_orb0  = w;                                 // tot*104
    float* acc      = ef_orb0 + (size_t)tot*LW;
    float* accSum   = acc     + (size_t)tot*LW;
    float* density  = accSum  + (size_t)tot*LW;          // tot*64
    float* coeff    = density + (size_t)tot*NORBIT;      // tot*8

    const int PB = 512;                                  // pair block (7680 % 512 == 0)
    size_t lds_bytes = ((size_t)2*natom*LW + (size_t)natom*NWAVE) * sizeof(float);
    bool use_lds = (maxneigh % PB == 0) && (lds_bytes <= 160*1024);

    dim3 blk(256);
    dim3 gAtom((tot   + 255)/256);
    dim3 gPairL((npair + PB - 1)/PB);
    dim3 gPairD((npair + 255)/256);
    dim3 gDens((tot/16 + 7)/8);                          // 1 wave / 16-atom tile

    k_init<<<gAtom, blk, 0, stream>>>(ef, ef_para, p_coeff, species,
                                      ef_orb0, acc, accSum, coeff, natom, tot);

    for (int t = 0; t < 3; ++t) {
        int use_sum = (t > 0) ? 1 : 0;
        if (use_lds) {
            k_pairs_lds<<<gPairL, dim3(PB), lds_bytes, stream>>>(
                cart, atom_idx, shifts, species, rs, inta,
                coeff, accSum, acc, natom, maxneigh, npair, use_sum);
        } else {
            k_pairs_direct<<<gPairD, blk, 0, stream>>>(
                cart, atom_idx, shifts, species, rs, inta,
                coeff, accSum, acc, natom, maxneigh, npair, use_sum);
        }
        float* ddst = (t == 2) ? (float*)d_out : density;
        k_density<<<gDens, blk, 0, stream>>>(acc, hyper, ddst, tot);
        if (t < 2) {
            k_update<<<gAtom, blk, 0, stream>>>(density,
                                                oc_w  + (size_t)t*NORBIT*NWAVE,
                                                oc_emb+ (size_t)t*NTYPE*NWAVE,
                                                species, ef_orb0,
                                                coeff, acc, accSum, tot);
        }
    }
}